// StructuredKernel_79439715107059
// MI455X (gfx1250) — compile-verified
//
#include <hip/hip_runtime.h>
#include <math.h>

// ---------------------------------------------------------------------------
// MMD loss between weighted 2-D point clouds, gated by a descriptor MLP.
// Pairwise x.y^T GEMM (K=2 padded to 4) runs on the CDNA5 matrix pipe via
// V_WMMA_F32_16X16X4_F32; exp/weighting co-executes on VALU.
// ---------------------------------------------------------------------------

#define N_PTS 8192
#define FEATN 5
#define HIDN  32
#define DESCN 4
#define COLT_PER_CHUNK 16   // column tiles handled per wave

typedef __attribute__((ext_vector_type(2))) float v2f;
typedef __attribute__((ext_vector_type(8))) float v8f;

__device__ __forceinline__ float softplus_f(float x) {
    // numerically stable log(1+e^x)
    return fmaxf(x, 0.0f) + log1pf(__expf(-fabsf(x)));
}

// ---------------------------------------------------------------------------
// Zero the small accumulator region of the workspace.
// acc[0..2] = base  {sum_w, sum_w*px, sum_w*py}
// acc[3..5] = target{sum_w, sum_w*px, sum_w*py}
// acc[6]    = loss accumulator
// ---------------------------------------------------------------------------
__global__ void mmd_zero_accums(float* acc) {
    int i = threadIdx.x;
    if (i < 7) acc[i] = 0.0f;
}

// ---------------------------------------------------------------------------
// Pass 1: per-point node-weight MLP (5->32->1, ReLU, softplus) and weighted
// sums for centering. Raw weight stashed in aux[i].y. n must be a multiple
// of blockDim.x (8192 % 256 == 0).
// ---------------------------------------------------------------------------
__global__ __launch_bounds__(256)
void mmd_weights_pass1(const float* __restrict__ feat,
                       const float* __restrict__ pos,
                       const float* __restrict__ w1,  // [5,32]
                       const float* __restrict__ b1,  // [32]
                       const float* __restrict__ w2,  // [32]
                       const float* __restrict__ b2,  // [1]
                       const float* __restrict__ log_scale, // [2]
                       float2* __restrict__ aux,      // aux[i].y <- raw weight
                       float* __restrict__ acc)       // {sum_w, swpx, swpy}
{
    __shared__ float s_red[3];
    if (threadIdx.x == 0) { s_red[0] = 0.f; s_red[1] = 0.f; s_red[2] = 0.f; }
    __syncthreads();

    const int i = blockIdx.x * blockDim.x + threadIdx.x;

    float f[FEATN];
#pragma unroll
    for (int k = 0; k < FEATN; ++k) f[k] = feat[i * FEATN + k];

    float logit = b2[0];
#pragma unroll 4
    for (int j = 0; j < HIDN; ++j) {
        float a = b1[j];
#pragma unroll
        for (int k = 0; k < FEATN; ++k) a = fmaf(f[k], w1[k * HIDN + j], a);
        logit = fmaf(fmaxf(a, 0.0f), w2[j], logit);
    }
    const float w = softplus_f(logit) + 1e-6f;

    float2 av; av.x = 0.0f; av.y = w;
    aux[i] = av;

    const float sx = __expf(log_scale[0]);
    const float sy = __expf(log_scale[1]);
    const float px = pos[2 * i + 0] * sx;
    const float py = pos[2 * i + 1] * sy;

    atomicAdd(&s_red[0], w);
    atomicAdd(&s_red[1], w * px);
    atomicAdd(&s_red[2], w * py);
    __syncthreads();
    if (threadIdx.x == 0) {
        atomicAdd(&acc[0], s_red[0]);
        atomicAdd(&acc[1], s_red[1]);
        atomicAdd(&acc[2], s_red[2]);
    }
}

// ---------------------------------------------------------------------------
// Pass 2: write centered scaled positions and {|p|^2, normalized w} aux.
// ---------------------------------------------------------------------------
__global__ __launch_bounds__(256)
void mmd_prep_pass2(const float* __restrict__ pos,
                    const float* __restrict__ log_scale,
                    float2* __restrict__ posc,
                    float2* __restrict__ aux,
                    const float* __restrict__ acc)
{
    const int i = blockIdx.x * blockDim.x + threadIdx.x;
    const float sw  = acc[0];
    const float mx  = acc[1] / sw;
    const float my  = acc[2] / sw;
    const float sx  = __expf(log_scale[0]);
    const float sy  = __expf(log_scale[1]);
    const float px  = pos[2 * i + 0] * sx - mx;
    const float py  = pos[2 * i + 1] * sy - my;

    float2 p; p.x = px; p.y = py;
    posc[i] = p;

    float2 a = aux[i];
    float2 o; o.x = px * px + py * py; o.y = a.y / sw;
    aux[i] = o;
}

// ---------------------------------------------------------------------------
// Pairwise MMD term: one wave per (16-row tile, chunk of 16 column tiles).
//   dot tile via V_WMMA_F32_16X16X4_F32 (K0=x, K1=y, K2=K3=0)
//   acc += wx[M] * wy[N] * exp((2*dot - |x|^2 - |y|^2) * inv2sig2)
// A layout (32-bit 16x4): lanes 0-15 -> {K0,K1} of row M=lane; lanes 16-31
// hold the K2/K3 half (zeros here). B mirrored. D VGPR r, lane l:
// M = r + 8*(l>=16), N = l&15.
// ---------------------------------------------------------------------------
__global__ __launch_bounds__(32)
void mmd_pairwise(const float2* __restrict__ posX,
                  const float2* __restrict__ auxX,
                  const float2* __restrict__ posY,
                  const float2* __restrict__ auxY,
                  const float* __restrict__ log_sigma,
                  float coef,
                  float* __restrict__ loss_acc)
{
    const int lane = threadIdx.x;
    const int hi   = lane >> 4;        // 0: K0/K1 half, 1: K2/K3 half (zeros)
    const int lo   = lane & 15;
    const int m0   = blockIdx.x * 16;

    const float sigma = __expf(log_sigma[0]);
    const float inv   = 1.0f / (2.0f * sigma * sigma);

    // A operand: fixed for the whole wave's row tile.
    v2f a;
    if (hi == 0) {
        float2 p = posX[m0 + lo];
        a[0] = p.x; a[1] = p.y;
    } else {
        a[0] = 0.0f; a[1] = 0.0f;
    }

    // Hoist row aux for the 8 D registers (M = r + 8*hi).
    float rowx2[8], roww[8];
#pragma unroll
    for (int r = 0; r < 8; ++r) {
        float2 t = auxX[m0 + r + hi * 8];
        rowx2[r] = t.x; roww[r] = t.y;
    }

    float accw = 0.0f;
    const int t0 = blockIdx.y * COLT_PER_CHUNK;

    for (int t = 0; t < COLT_PER_CHUNK; ++t) {
        const int n0 = (t0 + t) * 16;

        v2f b;
        if (hi == 0) {
            float2 p = posY[n0 + lo];
            b[0] = p.x; b[1] = p.y;
        } else {
            b[0] = 0.0f; b[1] = 0.0f;
        }
        const float2 ya = auxY[n0 + lo];   // {|y|^2, wy} for N = lo

        v8f c = {};
        v8f d = __builtin_amdgcn_wmma_f32_16x16x4_f32(
            /*neg_a=*/false, a, /*neg_b=*/false, b,
            /*c_mod=*/(short)0, c, /*reuse_a=*/false, /*reuse_b=*/false);

#pragma unroll
        for (int r = 0; r < 8; ++r) {
            const float arg = (2.0f * d[r] - rowx2[r] - ya.x) * inv;
            accw = fmaf(roww[r] * ya.y, __expf(arg), accw);
        }
    }

    // wave32 reduction
#pragma unroll
    for (int off = 16; off >= 1; off >>= 1)
        accw += __shfl_xor(accw, off, 32);
    if (lane == 0) atomicAdd(loss_acc, coef * accw);
}

// ---------------------------------------------------------------------------
// Finalize: descriptor gate MLP (4->32->1) and output scalar.
// ---------------------------------------------------------------------------
__global__ void mmd_finalize(const float* __restrict__ bd,
                             const float* __restrict__ td,
                             const float* __restrict__ g1,
                             const float* __restrict__ gb1,
                             const float* __restrict__ g2,
                             const float* __restrict__ gb2,
                             const float* __restrict__ bias,
                             const float* __restrict__ loss_acc,
                             float* __restrict__ out)
{
    if (threadIdx.x != 0 || blockIdx.x != 0) return;
    float diff[DESCN];
#pragma unroll
    for (int k = 0; k < DESCN; ++k) diff[k] = fabsf(bd[k] - td[k]);
    float logit = gb2[0];
    for (int j = 0; j < HIDN; ++j) {
        float a = gb1[j];
#pragma unroll
        for (int k = 0; k < DESCN; ++k) a = fmaf(diff[k], g1[k * HIDN + j], a);
        logit = fmaf(fmaxf(a, 0.0f), g2[j], logit);
    }
    const float gate = softplus_f(logit) + 1e-6f;
    out[0] = loss_acc[0] * gate + bias[0];
}

// ---------------------------------------------------------------------------
extern "C" void kernel_launch(void* const* d_in, const int* in_sizes, int n_in,
                              void* d_out, int out_size, void* d_ws, size_t ws_size,
                              hipStream_t stream) {
    const float* base_pos    = (const float*)d_in[0];
    const float* base_feat   = (const float*)d_in[1];
    const float* base_desc   = (const float*)d_in[2];
    const float* target_pos  = (const float*)d_in[3];
    const float* target_feat = (const float*)d_in[4];
    const float* target_desc = (const float*)d_in[5];
    const float* w1  = (const float*)d_in[6];
    const float* b1  = (const float*)d_in[7];
    const float* w2  = (const float*)d_in[8];
    const float* b2  = (const float*)d_in[9];
    const float* g1  = (const float*)d_in[10];
    const float* gb1 = (const float*)d_in[11];
    const float* g2  = (const float*)d_in[12];
    const float* gb2 = (const float*)d_in[13];
    const float* log_sigma = (const float*)d_in[14];
    const float* log_scale = (const float*)d_in[15];
    const float* bias      = (const float*)d_in[16];
    (void)in_sizes; (void)n_in; (void)out_size; (void)ws_size;

    float* ws = (float*)d_ws;
    float2* posc_b = (float2*)(ws);
    float2* aux_b  = (float2*)(ws + N_PTS * 2);
    float2* posc_t = (float2*)(ws + N_PTS * 4);
    float2* aux_t  = (float2*)(ws + N_PTS * 6);
    float*  acc    = ws + N_PTS * 8;   // 7 floats

    float* out = (float*)d_out;

    mmd_zero_accums<<<1, 32, 0, stream>>>(acc);

    const int nblk = N_PTS / 256;
    mmd_weights_pass1<<<nblk, 256, 0, stream>>>(base_feat, base_pos, w1, b1, w2, b2,
                                                log_scale, aux_b, &acc[0]);
    mmd_weights_pass1<<<nblk, 256, 0, stream>>>(target_feat, target_pos, w1, b1, w2, b2,
                                                log_scale, aux_t, &acc[3]);

    mmd_prep_pass2<<<nblk, 256, 0, stream>>>(base_pos, log_scale, posc_b, aux_b, &acc[0]);
    mmd_prep_pass2<<<nblk, 256, 0, stream>>>(target_pos, log_scale, posc_t, aux_t, &acc[3]);

    dim3 pgrid(N_PTS / 16, N_PTS / (16 * COLT_PER_CHUNK));
    float* loss = &acc[6];
    mmd_pairwise<<<pgrid, 32, 0, stream>>>(posc_b, aux_b, posc_b, aux_b, log_sigma,  1.0f, loss);
    mmd_pairwise<<<pgrid, 32, 0, stream>>>(posc_b, aux_b, posc_t, aux_t, log_sigma, -2.0f, loss);
    mmd_pairwise<<<pgrid, 32, 0, stream>>>(posc_t, aux_t, posc_t, aux_t, log_sigma,  1.0f, loss);

    mmd_finalize<<<1, 1, 0, stream>>>(base_desc, target_desc, g1, gb1, g2, gb2,
                                      bias, loss, out);
}